// AscendRejectionSampler_23081154249008
// MI455X (gfx1250) — compile-verified
//
#include <hip/hip_runtime.h>

// Rejection sampler for speculative decoding (B requests, SPEC draft tokens,
// vocab V). Memory-bound streaming argmax -> CDNA5 async global->LDS copy
// engine (ASYNCcnt), depth-4 rotating pipeline per wave, wave32 shuffle
// reductions. No matmul content, so no WMMA by design.

#define WAVES 8
#define CHUNK 128  // floats per wave-chunk: 32 lanes * 16B (b128) = 512B
#define DEPTH 4    // pipeline stages (buffer index = chunk_iter & 3)

__device__ __forceinline__ void async_copy16(unsigned lds_off, unsigned voff,
                                             const float* base) {
  // GVS mode: global address is the 64-bit saddr base plus the 32-bit vaddr offset; vdst holds the LDS address
  asm volatile("global_load_async_to_lds_b128 %0, %1, %2"
               :
               : "v"(lds_off), "v"(voff), "s"(base)
               : "memory");
}

__global__ __launch_bounds__(256) void rs_rowreduce(
    const int* __restrict__ draft_ids, const int* __restrict__ cu,
    const float* __restrict__ draft_probs, const float* __restrict__ target_probs,
    const float* __restrict__ uniform_probs, const float* __restrict__ qmat,
    const unsigned char* __restrict__ is_greedy, int B, int V,
    int* __restrict__ ws_argmax, int* __restrict__ ws_recov,
    int* __restrict__ ws_acc) {
  __shared__ __align__(16) float lds_t[DEPTH][WAVES][CHUNK];
  __shared__ __align__(16) float lds_d[DEPTH][WAVES][CHUNK];
  __shared__ __align__(16) float lds_q[DEPTH][WAVES][CHUNK];
  __shared__ float s_tv[WAVES];
  __shared__ int s_ti[WAVES];
  __shared__ float s_rv[WAVES];
  __shared__ int s_ri[WAVES];

  const int t = blockIdx.x;
  const int tid = threadIdx.x;
  const int w = tid >> 5;
  const int l = tid & 31;

  // request id: searchsorted(cu, t, side='right') via binary search (uniform,
  // 7 scalar-load steps instead of up to B dependent ones)
  int lo = 0, hi = B;
  while (lo < hi) {
    int mid = (lo + hi) >> 1;
    if (t >= cu[mid]) lo = mid + 1; else hi = mid;
  }
  const int req = lo;
  const bool greedy = is_greedy[req] != 0;

  const float* trow = target_probs + (size_t)t * (size_t)V;
  const float* drow = draft_probs + (size_t)t * (size_t)V;
  const float* qrow = qmat + (size_t)req * (size_t)V;

  const int NC = V / CHUNK;  // 32000/128 = 250 full chunks

  float bt = -INFINITY;
  int it = 0x7fffffff;
  float br = -INFINITY;
  int ir = 0x7fffffff;

  // per-stage LDS byte offsets for this lane's 16B slot
  unsigned lt[DEPTH], ld[DEPTH], lq[DEPTH];
#pragma unroll
  for (int s = 0; s < DEPTH; ++s) {
    lt[s] = (unsigned)(size_t)&lds_t[s][w][l * 4];
    ld[s] = (unsigned)(size_t)&lds_d[s][w][l * 4];
    lq[s] = (unsigned)(size_t)&lds_q[s][w][l * 4];
  }

  auto issue = [&](int buf, int c) {
    unsigned voff = (unsigned)c * (CHUNK * 4u) + (unsigned)(l * 16);
    async_copy16(lt[buf], voff, trow);
    if (!greedy) {
      async_copy16(ld[buf], voff, drow);
      async_copy16(lq[buf], voff, qrow);
    }
  };

  auto process = [&](int buf, int c) {
    const float4 tv = *reinterpret_cast<const float4*>(&lds_t[buf][w][l * 4]);
    float tvs[4] = {tv.x, tv.y, tv.z, tv.w};
    const int base = c * CHUNK + l * 4;
    if (!greedy) {
      const float4 dv = *reinterpret_cast<const float4*>(&lds_d[buf][w][l * 4]);
      const float4 qv = *reinterpret_cast<const float4*>(&lds_q[buf][w][l * 4]);
      float dvs[4] = {dv.x, dv.y, dv.z, dv.w};
      float qvs[4] = {qv.x, qv.y, qv.z, qv.w};
#pragma unroll
      for (int j = 0; j < 4; ++j) {
        int idx = base + j;
        if (tvs[j] > bt) { bt = tvs[j]; it = idx; }  // local idx increasing
        float adj = tvs[j] - dvs[j];
        adj = adj > 0.0f ? adj : 0.0f;
        float r = adj / qvs[j];
        if (r > br) { br = r; ir = idx; }
      }
    } else {
#pragma unroll
      for (int j = 0; j < 4; ++j) {
        int idx = base + j;
        if (tvs[j] > bt) { bt = tvs[j]; it = idx; }
      }
    }
  };

  // chunks owned by this wave: c_j = w + j*WAVES, j = 0..M-1
  const int M = (NC > w) ? ((NC - w + WAVES - 1) / WAVES) : 0;
  const int P = (M < DEPTH) ? M : DEPTH;

  // prologue: fill the pipeline
  for (int j = 0; j < P; ++j) issue(j & 3, w + j * WAVES);

  // steady state: wait for oldest stage (npend * (DEPTH-1) younger loads may
  // still be in flight; async loads complete in order), consume it, refill it.
  int j = 0;
  for (; j + DEPTH < M; ++j) {
    if (!greedy)
      asm volatile("s_wait_asynccnt 0x9" ::: "memory");  // 3 arrays * 3 stages
    else
      asm volatile("s_wait_asynccnt 0x3" ::: "memory");  // 1 array  * 3 stages
    process(j & 3, w + j * WAVES);
    // drain DScnt before the async engine overwrites this LDS buffer: DS and
    // ASYNC counters are mutually unordered, so in-flight ds_loads must retire
    // before the refill lands.
    asm volatile("s_wait_dscnt 0x0" ::: "memory");
    issue(j & 3, w + (j + DEPTH) * WAVES);
  }
  // drain: everything outstanding is ours; one full wait, then consume.
  asm volatile("s_wait_asynccnt 0x0" ::: "memory");
  for (; j < M; ++j) process(j & 3, w + j * WAVES);

  // generic tail (V not multiple of CHUNK): direct loads, all 256 threads
  for (int v = NC * CHUNK + tid; v < V; v += 256) {
    float tvv = trow[v];
    if (tvv > bt) { bt = tvv; it = v; }
    if (!greedy) {
      float adj = tvv - drow[v];
      adj = adj > 0.0f ? adj : 0.0f;
      float r = adj / qrow[v];
      if (r > br) { br = r; ir = v; }
    }
  }

  // wave32 shuffle reduction with first-index tie break (matches jnp.argmax)
#pragma unroll
  for (int off = 16; off > 0; off >>= 1) {
    float ov = __shfl_xor(bt, off, 32);
    int oi = __shfl_xor(it, off, 32);
    if (ov > bt || (ov == bt && oi < it)) { bt = ov; it = oi; }
    float orv = __shfl_xor(br, off, 32);
    int ori = __shfl_xor(ir, off, 32);
    if (orv > br || (orv == br && ori < ir)) { br = orv; ir = ori; }
  }
  if (l == 0) {
    s_tv[w] = bt; s_ti[w] = it;
    s_rv[w] = br; s_ri[w] = ir;
  }
  __syncthreads();

  if (tid == 0) {
    float ft = -INFINITY; int fti = 0x7fffffff;
    float fr = -INFINITY; int fri = 0x7fffffff;
#pragma unroll
    for (int i = 0; i < WAVES; ++i) {
      if (s_tv[i] > ft || (s_tv[i] == ft && s_ti[i] < fti)) { ft = s_tv[i]; fti = s_ti[i]; }
      if (s_rv[i] > fr || (s_rv[i] == fr && s_ri[i] < fri)) { fr = s_rv[i]; fri = s_ri[i]; }
    }
    ws_argmax[t] = fti;
    if (!greedy) {
      ws_recov[t] = fri;
      int did = draft_ids[t];
      float pd = drow[did];
      float pt = trow[did];
      float ratio = pt / (pd > 0.0f ? pd : 1.0f);
      ws_acc[t] = (pd > 0.0f && ratio >= uniform_probs[t]) ? 1 : 0;
    }
  }
}

__global__ void rs_finalize(const int* __restrict__ draft_ids,
                            const int* __restrict__ cu,
                            const int* __restrict__ bonus,
                            const unsigned char* __restrict__ is_greedy,
                            const int* __restrict__ ws_argmax,
                            const int* __restrict__ ws_recov,
                            const int* __restrict__ ws_acc, int B, int SPEC,
                            int* __restrict__ out) {
  int r = blockIdx.x * blockDim.x + threadIdx.x;
  if (r >= B) return;
  int end = cu[r];
  int start = (r > 0) ? cu[r - 1] : 0;
  int nd = end - start;
  int* row = out + (size_t)r * (SPEC + 1);
  for (int j = 0; j <= SPEC; ++j) row[j] = -1;  // PLACEHOLDER
  bool greedy = is_greedy[r] != 0;
  bool rej = false;
  for (int p = 0; p < nd && p <= SPEC; ++p) {
    int t = start + p;
    int am = ws_argmax[t];
    int dtk = draft_ids[t];
    bool acc = greedy ? (dtk == am) : (ws_acc[t] != 0);
    int val = greedy ? am : (acc ? dtk : ws_recov[t]);
    if (!rej) row[p] = val;  // first rejected token is still written
    if (!acc) rej = true;
  }
  if (!rej && nd <= SPEC) row[nd] = bonus[r];  // bonus when fully accepted
}

extern "C" void kernel_launch(void* const* d_in, const int* in_sizes, int n_in,
                              void* d_out, int out_size, void* d_ws,
                              size_t ws_size, hipStream_t stream) {
  const int* draft_ids = (const int*)d_in[0];
  const int* cu = (const int*)d_in[1];
  const float* draft_probs = (const float*)d_in[2];
  const float* target_probs = (const float*)d_in[3];
  const int* bonus = (const int*)d_in[4];
  const float* uniform = (const float*)d_in[5];
  const float* q = (const float*)d_in[6];
  const unsigned char* greedy = (const unsigned char*)d_in[7];  // jax bool -> 1 byte

  const int T = in_sizes[0];
  const int B = in_sizes[4];
  const int V = (int)((long long)in_sizes[2] / T);
  const int SPEC = T / B;

  int* ws_argmax = (int*)d_ws;
  int* ws_recov = ws_argmax + T;
  int* ws_acc = ws_recov + T;

  rs_rowreduce<<<T, 256, 0, stream>>>(draft_ids, cu, draft_probs, target_probs,
                                      uniform, q, greedy, B, V, ws_argmax,
                                      ws_recov, ws_acc);
  rs_finalize<<<(B + 127) / 128, 128, 0, stream>>>(
      draft_ids, cu, bonus, greedy, ws_argmax, ws_recov, ws_acc, B, SPEC,
      (int*)d_out);
}